// DepthAttentionResidual_5282809774278
// MI455X (gfx1250) — compile-verified
//
#include <hip/hip_runtime.h>
#include <math.h>

#define S_  16
#define B_  4
#define T_  1024
#define D_  2048
#define BT_ (B_ * T_)          // 4096 (b,t) rows

typedef __attribute__((ext_vector_type(2))) float v2f;
typedef __attribute__((ext_vector_type(4))) float v4f;
typedef __attribute__((ext_vector_type(8))) float v8f;

// ---------------------------------------------------------------------------
// Pass 1: per (b,t) compute dot(query, x_s) and sumsq(x_s) for all 16 sources
// via V_WMMA_F32_16X16X4_F32 (A = 16 sources x 4 K, B = query chunk broadcast
// into all 16 columns -> every output column holds the same 16 dots).
// Then RMS-scale + softmax over the 16 sources, weights -> workspace [S, BT].
// One workgroup per (b,t); 8 waves split the K = 2048 dimension.
// ---------------------------------------------------------------------------
__global__ __launch_bounds__(256)
void k_scores(const float* __restrict__ q,
              const float* __restrict__ src,
              float* __restrict__ wgt) {
  const int bt   = blockIdx.x;
  const int tid  = threadIdx.x;
  const int lane = tid & 31;
  const int wave = tid >> 5;      // 0..7
  const int m    = lane & 15;     // A-matrix row = source index
  const int kh   = lane >> 4;     // K half: lanes 0-15 -> K 0,1 ; 16-31 -> K 2,3

  __shared__ float s_dot[8][16];  // per-wave partial dot per source
  __shared__ float s_sq[8][16];   // per-wave partial sumsq per source
  __shared__ float s_sc[16];      // scores

  // Row base for source m at this (b,t): offset = (m*BT + bt)*D
  const float* __restrict__ aptr = src + ((long long)m * BT_ + bt) * D_ + kh * 2;
  const float* __restrict__ bptr = q + kh * 2;

  v8f acc = {0.f, 0.f, 0.f, 0.f, 0.f, 0.f, 0.f, 0.f};
  float sq = 0.f;

  const int k0 = wave * (D_ / 8);          // 256 K-elements per wave
  #pragma unroll 4
  for (int k = k0; k < k0 + D_ / 8; k += 4) {
    // A fragment: lane holds row m, K = k + kh*2 + {0,1} (single-use -> NT)
    v2f a = __builtin_nontemporal_load((const v2f*)(aptr + k));
    // B fragment: query broadcast to all N columns (query is hot -> temporal)
    v2f b = *(const v2f*)(bptr + k);
    sq += a.x * a.x + a.y * a.y;
    acc = __builtin_amdgcn_wmma_f32_16x16x4_f32(
        /*neg_a=*/false, a, /*neg_b=*/false, b,
        /*c_mod=*/(short)0, acc, /*reuse_a=*/false, /*reuse_b=*/false);
  }

  // sumsq of row m = lane m partial + lane m+16 partial
  sq += __shfl_xor(sq, 16, 32);
  if (lane < 16) s_sq[wave][m] = sq;

  // C layout (16x16 f32): lanes 0-15 hold M=0..7 in c[0..7], lanes 16-31 hold
  // M=8..15. All columns are identical, so lanes 0 and 16 suffice.
  if (lane == 0) {
    #pragma unroll
    for (int r = 0; r < 8; ++r) s_dot[wave][r] = acc[r];
  } else if (lane == 16) {
    #pragma unroll
    for (int r = 0; r < 8; ++r) s_dot[wave][8 + r] = acc[r];
  }
  __syncthreads();

  if (tid < 16) {
    float dot = 0.f, ssq = 0.f;
    #pragma unroll
    for (int w = 0; w < 8; ++w) { dot += s_dot[w][tid]; ssq += s_sq[w][tid]; }
    const float ms = ssq * (1.0f / D_);
    // score = dot(q,x) * rsqrt(mean(x^2)+eps) / sqrt(D)   (fp32 eps, T=1.0)
    s_sc[tid] = dot * rsqrtf(ms + 1.1920929e-07f) * 0.022097086912079608f;
  }
  __syncthreads();

  if (tid == 0) {                 // softmax over 16 sources (deterministic)
    float mx = s_sc[0];
    #pragma unroll
    for (int s = 1; s < 16; ++s) mx = fmaxf(mx, s_sc[s]);
    float e[16], sum = 0.f;
    #pragma unroll
    for (int s = 0; s < 16; ++s) { e[s] = __expf(s_sc[s] - mx); sum += e[s]; }
    const float inv = 1.f / sum;
    #pragma unroll
    for (int s = 0; s < 16; ++s) wgt[s * BT_ + bt] = e[s] * inv;
  }
}

// ---------------------------------------------------------------------------
// Pass 2: mixed[bt,d] = sum_s w[s,bt] * x[s,bt,d].  Pure streaming: float4
// nontemporal loads (single-use data), NT store, weights staged in LDS.
// Grid: (D/1024, BT), 256 threads x float4 = 1024 elements of d per block.
// ---------------------------------------------------------------------------
__global__ __launch_bounds__(256)
void k_mix(const float* __restrict__ src,
           const float* __restrict__ wgt,
           float* __restrict__ out) {
  const int bt  = blockIdx.y;
  const int tid = threadIdx.x;
  __shared__ float w[16];
  if (tid < 16) w[tid] = wgt[tid * BT_ + bt];
  __syncthreads();

  const int d0 = blockIdx.x * 1024 + tid * 4;
  const float* p = src + (long long)bt * D_ + d0;
  v4f acc = {0.f, 0.f, 0.f, 0.f};
  #pragma unroll
  for (int s = 0; s < 16; ++s) {
    v4f v = __builtin_nontemporal_load((const v4f*)(p + (long long)s * BT_ * D_));
    const float ws = w[s];
    acc.x += ws * v.x; acc.y += ws * v.y; acc.z += ws * v.z; acc.w += ws * v.w;
  }
  __builtin_nontemporal_store(acc, (v4f*)(out + (long long)bt * D_ + d0));
}

// ---------------------------------------------------------------------------
extern "C" void kernel_launch(void* const* d_in, const int* in_sizes, int n_in,
                              void* d_out, int out_size, void* d_ws, size_t ws_size,
                              hipStream_t stream) {
  const float* q   = (const float*)d_in[0];   // [D] fp32
  const float* src = (const float*)d_in[1];   // [S,B,T,D] fp32
  // d_in[2] = num_sources scalar (compile-time S_ = 16)
  float* out = (float*)d_out;                 // [B,T,D] fp32
  float* wgt = (float*)d_ws;                  // [S, B*T] fp32 = 256 KB scratch

  k_scores<<<BT_, 256, 0, stream>>>(q, src, wgt);
  k_mix<<<dim3(D_ / 1024, BT_), 256, 0, stream>>>(src, wgt, out);
}